// LevelInit_39651138077360
// MI455X (gfx1250) — compile-verified
//
#include <hip/hip_runtime.h>

typedef __attribute__((ext_vector_type(2))) float v2f;
typedef __attribute__((ext_vector_type(8))) float v8f;

#define B_   2
#define CIN  3
#define HIN  384
#define WIN  1280
#define HQ   96
#define WL   320
#define WR   1280
#define DMAX 64

__device__ __forceinline__ float leaky(float x) { return x >= 0.f ? x : 0.2f * x; }

// ---------------------------------------------------------------------------
// Kernel 1: fused [4x4 reduce conv (WMMA, K=48) -> leaky -> 1x1 em conv
// (WMMA, K=16) -> leaky].  One wave owns 16 output pixels x 16 channels.
//   Phase 0: im2col patch (48 x 16) staged in LDS (branchless: clamped
//            address + select for the SAME-padding border).
//   Phase 1: reduce conv = 16x16x48 GEMM via 12x V_WMMA_F32_16X16X4_F32,
//            A = w_reduce (OIHW flat = [M][k], k = ci*16+ky*4+kx).
//            leaky(C) bounced through LDS to convert C-layout -> B-layout.
//   Phase 2: em conv = 16x16x16 GEMM via 4x WMMA, leaky, store.
// Lane layouts (ISA 7.12.2, fp32):
//   A: lane L holds A[M=L&15][K = 4t + (L<16 ? {0,1} : {2,3})]
//   B: lane L holds B[K = 4t + (L<16 ? {0,1} : {2,3})][N=L&15]
//   C: VGPR r <-> M = r + (L<16 ? 0 : 8), N = L&15.
// ---------------------------------------------------------------------------
template <int RIGHT, int WOUT>
__global__ void fused_reduce_em(const float* __restrict__ in,
                                const float* __restrict__ w_reduce,
                                const float* __restrict__ b_reduce,
                                const float* __restrict__ w_em,
                                const float* __restrict__ b_em,
                                float* __restrict__ out) {
  __shared__ float patch[8][48 * 16];      // per-wave im2col tile
  __shared__ float ldsR[8][16 * 16];       // per-wave leaky(reduce) tile
  const int wave = threadIdx.x >> 5;
  const int lane = threadIdx.x & 31;
  const int wpb  = blockDim.x >> 5;
  const int y    = blockIdx.y;
  const int b    = blockIdx.z;
  const int xs   = (blockIdx.x * wpb + wave) * 16;

  // ---- Phase 0: im2col into LDS (768/32 = 24 uniform iterations) ----
  float* P = patch[wave];
  for (int i = lane; i < 48 * 16; i += 32) {
    const int k  = i >> 4;                 // k = ci*16 + ky*4 + kx
    const int p  = i & 15;
    const int ci = k >> 4;
    const int ky = (k >> 2) & 3;
    const int kx = k & 3;
    const int xi = RIGHT ? (xs + p - 1 + kx) : ((xs + p) * 4 + kx);
    // Branchless border: unconditional load at clamped address + select.
    const int   xc  = xi < 0 ? 0 : (xi > WIN - 1 ? WIN - 1 : xi);
    const float raw = in[((b * CIN + ci) * HIN + (y * 4 + ky)) * WIN + xc];
    P[i] = (!RIGHT || xi == xc) ? raw : 0.f;
  }
  __syncthreads();

  // ---- Phase 1: reduce conv, 12 chained K=4 WMMAs ----
  const int hi = lane >= 16;
  const int N  = lane & 15;                // A-row M == B-col N for this lane
  const int M0 = hi ? 8 : 0;
  v8f c8;
#pragma unroll
  for (int r = 0; r < 8; ++r) c8[r] = b_reduce[M0 + r];
#pragma unroll
  for (int t = 0; t < 12; ++t) {
    const int kb = 4 * t + (hi ? 2 : 0);
    v2f a, bb;
    a.x  = w_reduce[N * 48 + kb];          // OIHW flat: [oc][k]
    a.y  = w_reduce[N * 48 + kb + 1];
    bb.x = P[kb * 16 + N];
    bb.y = P[(kb + 1) * 16 + N];
    c8 = __builtin_amdgcn_wmma_f32_16x16x4_f32(false, a, false, bb,
                                               (short)0, c8, false, false);
  }
  // C-layout -> B-layout conversion through LDS, with leaky.
  float* R = ldsR[wave];
#pragma unroll
  for (int r = 0; r < 8; ++r) R[(M0 + r) * 16 + N] = leaky(c8[r]);
  __syncthreads();

  // ---- Phase 2: em 1x1 conv, 4 chained K=4 WMMAs ----
#pragma unroll
  for (int r = 0; r < 8; ++r) c8[r] = b_em[M0 + r];
#pragma unroll
  for (int t = 0; t < 4; ++t) {
    const int kb = 4 * t + (hi ? 2 : 0);
    v2f a, bb;
    a.x  = w_em[N * 16 + kb];
    a.y  = w_em[N * 16 + kb + 1];
    bb.x = R[kb * 16 + N];
    bb.y = R[(kb + 1) * 16 + N];
    c8 = __builtin_amdgcn_wmma_f32_16x16x4_f32(false, a, false, bb,
                                               (short)0, c8, false, false);
  }
#pragma unroll
  for (int r = 0; r < 8; ++r)
    out[((b * 16 + M0 + r) * HQ + y) * WOUT + xs + N] = leaky(c8[r]);
}

// ---------------------------------------------------------------------------
// Kernel 2: cost volume + running min/argmin over disparity.
// One block per (y,b): full rt row (16x1280 = 80 KB) staged in dynamic LDS,
// 16 lt values in registers, cv streamed out coalesced along x.
// ---------------------------------------------------------------------------
__global__ void cost_volume(const float* __restrict__ lt,
                            const float* __restrict__ rt,
                            float* __restrict__ cv,      // [B,64,96,320]
                            float* __restrict__ cvmin,   // [B,96,320]
                            float* __restrict__ dmin) {  // [B,96,320]
  extern __shared__ float rrow[];          // 16*1280 floats
  const int y = blockIdx.x;
  const int b = blockIdx.y;
  const int x = threadIdx.x;               // 320 threads
  for (int i = threadIdx.x; i < 16 * WR; i += blockDim.x) {
    int c = i / WR, xw = i - c * WR;
    rrow[i] = rt[((b * 16 + c) * HQ + y) * WR + xw];
  }
  __syncthreads();

  float ltr[16];
#pragma unroll
  for (int c = 0; c < 16; ++c)
    ltr[c] = lt[((b * 16 + c) * HQ + y) * WL + x];

  float best  = 3.4e38f;
  int   bestd = 0;
  for (int d = 0; d < DMAX; ++d) {
    int xr = 4 * x + 1 - d;
    xr = xr < 0 ? 0 : (xr > WR - 1 ? WR - 1 : xr);
    float s = 0.f;
#pragma unroll
    for (int c = 0; c < 16; ++c) s += fabsf(ltr[c] - rrow[c * WR + xr]);
    cv[((b * DMAX + d) * HQ + y) * WL + x] = s;
    if (s < best) { best = s; bestd = d; }  // strict < => first-index argmin
  }
  cvmin[(b * HQ + y) * WL + x] = best;
  dmin [(b * HQ + y) * WL + x] = (float)bestd;
}

// ---------------------------------------------------------------------------
// Kernel 3: hypothesis head. 1x1 conv 17->13 as a padded 16x16x20 WMMA GEMM.
// Zero-padded A (16x20) is staged ONCE per block in LDS so the WMMA loop has
// unconditional ds_loads (no exec-masked global loads in the hot chain).
// Then pack p = [argmin, 0, 0, leaky(conv)].
// ---------------------------------------------------------------------------
__global__ void hyp_head(const float* __restrict__ lt,
                         const float* __restrict__ cvmin,
                         const float* __restrict__ dmin,
                         const float* __restrict__ w_hyp,
                         const float* __restrict__ b_hyp,
                         float* __restrict__ p) {        // [B,16,96,320]
  __shared__ float Aw[16 * 20];            // padded w_hyp: rows 13..15, K 17..19 = 0
  __shared__ float Bm[5][20 * 16];         // per-wave padded B operand
  const int wave = threadIdx.x >> 5;
  const int lane = threadIdx.x & 31;
  const int y    = blockIdx.y;
  const int b    = blockIdx.z;
  const int xs   = (blockIdx.x * 5 + wave) * 16;

  // Stage padded A in LDS (320/160 = 2 uniform iterations per thread).
  for (int i = threadIdx.x; i < 16 * 20; i += 160) {
    const int row = i / 20, k = i - row * 20;
    // Clamped unconditional load + select keeps this branch-free.
    const int   ic  = (row < 13 ? row : 12) * 17 + (k < 17 ? k : 16);
    const float raw = w_hyp[ic];
    Aw[i] = (row < 13 && k < 17) ? raw : 0.f;
  }

  // Fill B[K][N]: K=1..16 -> lt channels (uniform 8-iteration loop),
  // K=0 -> cv_min and K=17..19 -> 0 (lanes 0..15, re-converges before WMMA).
  float* Bw = Bm[wave];
  for (int i = lane; i < 16 * 16; i += 32) {
    const int k = (i >> 4) + 1, pp = i & 15;
    Bw[k * 16 + pp] = lt[((b * 16 + (k - 1)) * HQ + y) * WL + xs + pp];
  }
  if (lane < 16) {
    Bw[0 * 16 + lane]  = cvmin[(b * HQ + y) * WL + xs + lane];
    Bw[17 * 16 + lane] = 0.f;
    Bw[18 * 16 + lane] = 0.f;
    Bw[19 * 16 + lane] = 0.f;
  }
  __syncthreads();

  const int hi = lane >= 16;
  const int N  = lane & 15;
  const int M0 = hi ? 8 : 0;
  v8f c8;
#pragma unroll
  for (int r = 0; r < 8; ++r) {
    const int M = M0 + r;
    c8[r] = (M < 13) ? b_hyp[M] : 0.f;     // scalar-uniform per half-wave
  }
#pragma unroll
  for (int t = 0; t < 5; ++t) {
    const int kb = 4 * t + (hi ? 2 : 0);
    v2f a, bb;
    a.x  = Aw[N * 20 + kb];                // unconditional ds_load
    a.y  = Aw[N * 20 + kb + 1];
    bb.x = Bw[kb * 16 + N];
    bb.y = Bw[(kb + 1) * 16 + N];
    c8 = __builtin_amdgcn_wmma_f32_16x16x4_f32(false, a, false, bb,
                                               (short)0, c8, false, false);
  }
#pragma unroll
  for (int r = 0; r < 8; ++r) {
    const int M = M0 + r;
    if (M < 13)
      p[((b * 16 + 3 + M) * HQ + y) * WL + xs + N] = leaky(c8[r]);
  }
  if (lane < 16) {
    const int x = xs + lane;
    p[((b * 16 + 0) * HQ + y) * WL + x] = dmin[(b * HQ + y) * WL + x];
    p[((b * 16 + 1) * HQ + y) * WL + x] = 0.f;
    p[((b * 16 + 2) * HQ + y) * WL + x] = 0.f;
  }
}

// ---------------------------------------------------------------------------
extern "C" void kernel_launch(void* const* d_in, const int* in_sizes, int n_in,
                              void* d_out, int out_size, void* d_ws, size_t ws_size,
                              hipStream_t stream) {
  const float* l        = (const float*)d_in[0];
  const float* r        = (const float*)d_in[1];
  const float* w_reduce = (const float*)d_in[2];
  const float* b_reduce = (const float*)d_in[3];
  const float* w_em     = (const float*)d_in[4];
  const float* b_em     = (const float*)d_in[5];
  const float* w_hyp    = (const float*)d_in[6];
  const float* b_hyp    = (const float*)d_in[7];
  // d_in[8] = max_disp (64), compile-time constant here.

  float* p_out  = (float*)d_out;                       // [2,16,96,320]
  float* cv_out = p_out + (size_t)B_ * 16 * HQ * WL;   // [2,64,96,320]

  float* ws    = (float*)d_ws;
  float* rt    = ws;                                   // 2*16*96*1280
  float* lt    = rt + (size_t)B_ * 16 * HQ * WR;       // 2*16*96*320
  float* cvmin = lt + (size_t)B_ * 16 * HQ * WL;       // 2*96*320
  float* dmin  = cvmin + (size_t)B_ * HQ * WL;         // 2*96*320

  // Right branch: 1280 px/row, 8 waves * 16 px = 128 px per block.
  fused_reduce_em<1, WR><<<dim3(WR / 128, HQ, B_), 256, 0, stream>>>(
      r, w_reduce, b_reduce, w_em, b_em, rt);
  // Left branch: 320 px/row, 4 waves * 16 px = 64 px per block.
  fused_reduce_em<0, WL><<<dim3(WL / 64, HQ, B_), 128, 0, stream>>>(
      l, w_reduce, b_reduce, w_em, b_em, lt);
  // Cost volume: one block per (y,b), rt row staged in 80 KB dynamic LDS.
  cost_volume<<<dim3(HQ, B_), WL, 16 * WR * sizeof(float), stream>>>(
      lt, rt, cv_out, cvmin, dmin);
  // Hyp head: 5 waves * 16 px = 80 px per block.
  hyp_head<<<dim3(WL / 80, HQ, B_), 160, 0, stream>>>(
      lt, cvmin, dmin, w_hyp, b_hyp, p_out);
}